// GIN_84121229460233
// MI455X (gfx1250) — compile-verified
//
#include <hip/hip_runtime.h>
#include <hip/hip_bf16.h>

#define N_NODES 50000
#define N_EDGES 800000
#define DIM     128
#define MTILES_PER_BLOCK 8

typedef __attribute__((ext_vector_type(2))) float v2f;
typedef __attribute__((ext_vector_type(8))) float v8f;

// ---------------------------------------------------------------------------
// agg = src_feat (simple float4 copy, grid-exact)
// ---------------------------------------------------------------------------
__global__ void gin_copy_f4(const float* __restrict__ src,
                            float* __restrict__ dst, int n4) {
    int i = blockIdx.x * blockDim.x + threadIdx.x;
    if (i < n4) {
        reinterpret_cast<float4*>(dst)[i] =
            reinterpret_cast<const float4*>(src)[i];
    }
}

// ---------------------------------------------------------------------------
// agg[dst[e]] += h[src[e]]  -- one wave32 per edge, float4 per lane
// (coalesced 512B gather, 4 global fp32 atomic adds per lane)
// ---------------------------------------------------------------------------
__global__ void gin_scatter_add(const float* __restrict__ h,
                                const int* __restrict__ src,
                                const int* __restrict__ dst,
                                float* __restrict__ agg, int nEdges) {
    int gid  = blockIdx.x * blockDim.x + threadIdx.x;
    int e    = gid >> 5;
    int lane = gid & 31;
    if (e >= nEdges) return;

    int s = src[e];
    int d = dst[e];

    const float4 v = *reinterpret_cast<const float4*>(
        h + (size_t)s * DIM + lane * 4);
    float* base = agg + (size_t)d * DIM + lane * 4;
    atomicAdd(base + 0, v.x);
    atomicAdd(base + 1, v.y);
    atomicAdd(base + 2, v.z);
    atomicAdd(base + 3, v.w);
}

// ---------------------------------------------------------------------------
// out = relu(in @ w + bias), in:[nRows,128] w:[128,128] row-major.
// Block = 256 threads = 8 waves. Weights staged to LDS once per block,
// then each wave hoists its 16-col B fragments into 64 VGPRs and loops
// over MTILES_PER_BLOCK 16-row M-tiles: steady state is pure
// global_load_b64 (A) + v_wmma_f32_16x16x4_f32.
// ---------------------------------------------------------------------------
__global__ __launch_bounds__(256) void
gin_gemm_bias_relu(const float* __restrict__ in,
                   const float* __restrict__ w,
                   const float* __restrict__ bias,
                   float* __restrict__ out, int nRows) {
    __shared__ float lds_w[DIM * DIM];   // 65536 bytes

    // cooperative weight stage: 4096 float4 / 256 threads = 16 each
    {
        const float4* wsrc = reinterpret_cast<const float4*>(w);
        float4*       wdst = reinterpret_cast<float4*>(lds_w);
        for (int i = threadIdx.x; i < (DIM * DIM) / 4; i += 256)
            wdst[i] = wsrc[i];
    }
    __syncthreads();

    const int wave  = threadIdx.x >> 5;      // 0..7  -> N tile
    const int lane  = threadIdx.x & 31;
    const int lrow  = lane & 15;             // M (for A) / N (for B) index
    const int hi    = lane >> 4;             // 0: K={0,1}, 1: K={2,3}
    const int nbase = wave * 16;
    const int col   = nbase + lrow;

    // Hoist B fragments for this wave's N-tile into registers (one-time):
    // lane (lrow, hi), step kk: w[4kk+2hi][col], w[4kk+2hi+1][col]
    const float* bcol = lds_w + col;
    v2f bfrag[DIM / 4];
#pragma unroll
    for (int kk = 0; kk < DIM / 4; ++kk) {
        bfrag[kk].x = bcol[(4 * kk + 2 * hi + 0) * DIM];
        bfrag[kk].y = bcol[(4 * kk + 2 * hi + 1) * DIM];
    }
    const float bv = bias[col];

    for (int t = 0; t < MTILES_PER_BLOCK; ++t) {
        const int mbase = (blockIdx.x * MTILES_PER_BLOCK + t) * 16;
        if (mbase >= nRows) break;

        // A: lane (lrow, hi) holds in[mbase+lrow][k+2hi .. k+2hi+1]
        const float* arow = in + (size_t)(mbase + lrow) * DIM;

        v8f acc = {};
#pragma unroll
        for (int kk = 0; kk < DIM / 4; ++kk) {
            v2f a = *reinterpret_cast<const v2f*>(arow + 4 * kk + 2 * hi);
            acc = __builtin_amdgcn_wmma_f32_16x16x4_f32(
                /*neg_a=*/false, a, /*neg_b=*/false, bfrag[kk],
                /*c_mod=*/(short)0, acc, /*reuse_a=*/false, /*reuse_b=*/false);
        }

        // Epilogue: C/D layout -> VGPR r, lane l: row = r+8*(l>=16), col = l%16
#pragma unroll
        for (int r = 0; r < 8; ++r) {
            int   row = mbase + r + 8 * hi;
            float v   = acc[r] + bv;
            out[(size_t)row * DIM + col] = v > 0.f ? v : 0.f;
        }
    }
}

// ---------------------------------------------------------------------------
extern "C" void kernel_launch(void* const* d_in, const int* in_sizes, int n_in,
                              void* d_out, int out_size, void* d_ws,
                              size_t ws_size, hipStream_t stream) {
    const float* features = (const float*)d_in[0];
    const int*   src      = (const int*)d_in[1];
    const int*   dst      = (const int*)d_in[2];
    const float* w1a = (const float*)d_in[3];
    const float* b1a = (const float*)d_in[4];
    const float* w2a = (const float*)d_in[5];
    const float* b2a = (const float*)d_in[6];
    const float* w1b = (const float*)d_in[7];
    const float* b1b = (const float*)d_in[8];
    const float* w2b = (const float*)d_in[9];
    const float* b2b = (const float*)d_in[10];

    float* ws0 = (float*)d_ws;                       // agg
    float* ws1 = ws0 + (size_t)N_NODES * DIM;        // tmp
    float* ws2 = ws1 + (size_t)N_NODES * DIM;        // h after layer A

    const int n4       = (N_NODES * DIM) / 4;        // 1,600,000 float4
    const int copyBlks = (n4 + 255) / 256;
    const int scatBlks = ((N_EDGES * 32) + 255) / 256;
    const int mTiles   = N_NODES / 16;               // 3125, exact
    const int gemmBlks = (mTiles + MTILES_PER_BLOCK - 1) / MTILES_PER_BLOCK;

    // ---- Layer A ----
    gin_copy_f4<<<copyBlks, 256, 0, stream>>>(features, ws0, n4);
    gin_scatter_add<<<scatBlks, 256, 0, stream>>>(features, src, dst, ws0,
                                                  N_EDGES);
    gin_gemm_bias_relu<<<gemmBlks, 256, 0, stream>>>(ws0, w1a, b1a, ws1,
                                                     N_NODES);
    gin_gemm_bias_relu<<<gemmBlks, 256, 0, stream>>>(ws1, w2a, b2a, ws2,
                                                     N_NODES);

    // ---- Layer B ----
    gin_copy_f4<<<copyBlks, 256, 0, stream>>>(ws2, ws0, n4);
    gin_scatter_add<<<scatBlks, 256, 0, stream>>>(ws2, src, dst, ws0, N_EDGES);
    gin_gemm_bias_relu<<<gemmBlks, 256, 0, stream>>>(ws0, w1b, b1b, ws1,
                                                     N_NODES);
    gin_gemm_bias_relu<<<gemmBlks, 256, 0, stream>>>(ws1, w2b, b2b,
                                                     (float*)d_out, N_NODES);
}